// PartPrototypeBank_48009144435258
// MI455X (gfx1250) — compile-verified
//
#include <hip/hip_runtime.h>
#include <hip/hip_bf16.h>

// ---------------------------------------------------------------------------
// PartPrototypeBank for MI455X (gfx1250, wave32, WMMA + async global->LDS)
// Outputs (flat concat): new_part_bank [20000*8*256], new_embed_bank
// [20000*512], nce_loss [1], align_loss [1].
// ---------------------------------------------------------------------------

typedef float v2f __attribute__((ext_vector_type(2)));
typedef float v8f __attribute__((ext_vector_type(8)));

constexpr int   BATCH    = 512;
constexpr int   KPART    = 8;
constexpr int   DPART    = 256;
constexpr int   DE       = 512;
constexpr int   CLS      = 20000;
constexpr int   PROW     = KPART * DPART;            // 2048
constexpr long long P_ELEMS = (long long)CLS * PROW; // 40,960,000
constexpr long long E_ELEMS = (long long)CLS * DE;   // 10,240,000
constexpr float MOM      = 0.999f;
constexpr float INV_TEMP = 1.0f / 0.07f;
constexpr int   NHARD    = 16;
constexpr int   LDP      = 36;  // padded LDS row pitch (floats): 16B-aligned,
                                // 36*n mod 64 distinct over 16 lanes

__device__ __forceinline__ void wait_async0() {
#if __has_builtin(__builtin_amdgcn_s_wait_asynccnt)
  __builtin_amdgcn_s_wait_asynccnt(0);
#else
  asm volatile("s_wait_asynccnt 0x0" ::: "memory");
#endif
}

// ---------------- bulk bank copy (float4 streaming) ------------------------
__global__ void copy_f4_kernel(const float4* __restrict__ src,
                               float4* __restrict__ dst, long long n4) {
  long long i = (long long)blockIdx.x * blockDim.x + threadIdx.x;
  long long stride = (long long)gridDim.x * blockDim.x;
  for (; i < n4; i += stride) dst[i] = src[i];
}

// ---------------- EMA scatter update (one block per batch row) -------------
__global__ void ema_update_kernel(const float* __restrict__ part_features,
                                  const float* __restrict__ embeddings,
                                  const float* __restrict__ part_bank,
                                  const float* __restrict__ embed_bank,
                                  const int* __restrict__ labels,
                                  float* __restrict__ out_part,
                                  float* __restrict__ out_embed) {
  const int b = blockIdx.x;
  const int row = labels[b];
  const long long pbase = (long long)row * PROW;
  const long long fbase = (long long)b * PROW;
  for (int i = threadIdx.x; i < PROW; i += blockDim.x)
    out_part[pbase + i] = MOM * part_bank[pbase + i] + (1.0f - MOM) * part_features[fbase + i];
  const long long ebase = (long long)row * DE;
  const long long qbase = (long long)b * DE;
  for (int i = threadIdx.x; i < DE; i += blockDim.x)
    out_embed[ebase + i] = MOM * embed_bank[ebase + i] + (1.0f - MOM) * embeddings[qbase + i];
}

// ---------------- row normalize (one block per row) ------------------------
__global__ void row_normalize_kernel(const float* __restrict__ x,
                                     float* __restrict__ y, int ncols) {
  __shared__ float s[256];
  const int row = blockIdx.x;
  const long long base = (long long)row * ncols;
  float acc = 0.0f;
  for (int i = threadIdx.x; i < ncols; i += 256) {
    float v = x[base + i];
    acc += v * v;
  }
  s[threadIdx.x] = acc;
  __syncthreads();
  for (int st = 128; st >= 1; st >>= 1) {
    if (threadIdx.x < st) s[threadIdx.x] += s[threadIdx.x + st];
    __syncthreads();
  }
  const float inv = 1.0f / fmaxf(sqrtf(s[0]), 1e-12f);
  for (int i = threadIdx.x; i < ncols; i += 256) y[base + i] = x[base + i] * inv;
}

// ---------------- per-class inverse norms of updated embed bank ------------
__global__ void proto_invnorm_kernel(const float* __restrict__ eb,
                                     float* __restrict__ invn) {
  __shared__ float s[256];
  const int row = blockIdx.x;
  const long long base = (long long)row * DE;
  float acc = 0.0f;
  for (int i = threadIdx.x; i < DE; i += 256) {
    float v = eb[base + i];
    acc += v * v;
  }
  s[threadIdx.x] = acc;
  __syncthreads();
  for (int st = 128; st >= 1; st >>= 1) {
    if (threadIdx.x < st) s[threadIdx.x] += s[threadIdx.x + st];
    __syncthreads();
  }
  if (threadIdx.x == 0) invn[row] = 1.0f / fmaxf(sqrtf(s[0]), 1e-12f);
}

// ---------------- WMMA GEMM: sim = (q_hat @ eb^T) * invn * 1/TEMP ----------
// Block = 128 threads (4 waves). Block tile: 16 (M) x 64 (N), K chunked by 32.
// A tile staged with vector float4 stores; B tile staged with CDNA5 async
// global->LDS B128 (ASYNCcnt). Each wave: 8 x V_WMMA_F32_16X16X4_F32 / chunk.
__global__ void sim_gemm_wmma_kernel(const float* __restrict__ qn,
                                     const float* __restrict__ eb,
                                     const float* __restrict__ invn,
                                     float* __restrict__ sim) {
  __shared__ __align__(16) float As[16 * LDP];
  __shared__ __align__(16) float Bs[64 * LDP];

  const int tid  = threadIdx.x;
  const int lane = tid & 31;
  const int w    = tid >> 5;   // wave 0..3
  const int half = lane >> 4;  // 0: lanes 0-15, 1: lanes 16-31
  const int l16  = lane & 15;
  const int m0   = blockIdx.y * 16;
  const int nb   = blockIdx.x * 64;

  // --- static staging assignments (no loops, no guards) ---
  // A tile: 16 rows x 32 k = 128 float4; thread t owns one.
  const int ar  = tid >> 3;        // 0..15
  const int ac4 = tid & 7;         // 0..7  (float4 index along k)
  // B tile: 64 rows x 32 k = 512 float4; thread t owns 4 (t, t+128, ...).
  // Out-of-range class rows are clamped (their columns are never stored).
  const uint64_t eb_base = (uint64_t)eb;
  uint32_t lds_b[4];
  uint32_t goff_b[4];
#pragma unroll
  for (int i = 0; i < 4; ++i) {
    const int e  = tid + i * 128;
    const int n  = e >> 3;  // 0..63
    const int c4 = e & 7;   // 0..7
    int g = nb + n;
    g = (g < CLS) ? g : (CLS - 1);
    lds_b[i]  = (uint32_t)(uintptr_t)(&Bs[n * LDP + c4 * 4]);
    goff_b[i] = (uint32_t)((g * DE + c4 * 4) * sizeof(float));  // + k0*4 later
  }

  v8f c = {};
  for (int k0 = 0; k0 < DE; k0 += 32) {
    // A tile (vector LDS store)
    const float4 av =
        *(const float4*)&qn[(long long)(m0 + ar) * DE + k0 + ac4 * 4];
    *(float4*)&As[ar * LDP + ac4 * 4] = av;
    // B tile via async global->LDS (B128 per lane, GVS addressing)
    const uint32_t kbyte = (uint32_t)(k0 * sizeof(float));
#pragma unroll
    for (int i = 0; i < 4; ++i) {
      const uint32_t go = goff_b[i] + kbyte;
      asm volatile("global_load_async_to_lds_b128 %0, %1, %2"
                   :: "v"(lds_b[i]), "v"(go), "s"(eb_base)
                   : "memory");
    }
    wait_async0();
    __syncthreads();

#pragma unroll
    for (int kk = 0; kk < 32; kk += 4) {
      // A 16x4 f32: VGPR0 = K (lanes 0-15) / K+2 (lanes 16-31), VGPR1 = +1
      v2f a, bfrag;
      a.x = As[l16 * LDP + kk + half * 2 + 0];
      a.y = As[l16 * LDP + kk + half * 2 + 1];
      // B 4x16 f32: VGPR0 = row K (lanes 0-15) / row K+2 (lanes 16-31)
      bfrag.x = Bs[(w * 16 + l16) * LDP + kk + half * 2 + 0];
      bfrag.y = Bs[(w * 16 + l16) * LDP + kk + half * 2 + 1];
      c = __builtin_amdgcn_wmma_f32_16x16x4_f32(false, a, false, bfrag,
                                                (short)0, c, false, false);
    }
    __syncthreads();
  }
  // epilogue: C/D layout VGPR i -> row (i + 8*half), col l16
  const int n = nb + w * 16 + l16;
  if (n < CLS) {
    const float scale = invn[n] * INV_TEMP;
#pragma unroll
    for (int i = 0; i < 8; ++i) {
      const int mrow = m0 + i + half * 8;
      sim[(long long)mrow * CLS + n] = c[i] * scale;
    }
  }
}

// ---------------- top-16 + logsumexp per row (deterministic tree merge) ----
__device__ __forceinline__ void merge16(const float* a, const float* b, float* o) {
  int i = 0, j = 0;
#pragma unroll
  for (int k = 0; k < NHARD; ++k) o[k] = (a[i] >= b[j]) ? a[i++] : b[j++];
}

__global__ void topk_lse_kernel(const float* __restrict__ sim,
                                const int* __restrict__ labels,
                                float* __restrict__ nce_r) {
  __shared__ float cand[256 * NHARD];
  const int b = blockIdx.x;
  const int tid = threadIdx.x;
  const int label = labels[b];
  const long long base = (long long)b * CLS;

  float top[NHARD];
#pragma unroll
  for (int i = 0; i < NHARD; ++i) top[i] = -1e30f;

  for (int j = tid; j < CLS; j += 256) {
    if (j == label) continue;  // positive excluded from hard negatives
    const float v = sim[base + j];
    if (v > top[NHARD - 1]) {
      int p = NHARD - 1;
      while (p > 0 && v > top[p - 1]) { top[p] = top[p - 1]; --p; }
      top[p] = v;
    }
  }
#pragma unroll
  for (int i = 0; i < NHARD; ++i) cand[tid * NHARD + i] = top[i];
  __syncthreads();

  for (int st = 128; st >= 1; st >>= 1) {
    if (tid < st) {
      float o[NHARD];
      merge16(&cand[tid * NHARD], &cand[(tid + st) * NHARD], o);
      __syncthreads();
#pragma unroll
      for (int i = 0; i < NHARD; ++i) cand[tid * NHARD + i] = o[i];
    } else {
      __syncthreads();
    }
    __syncthreads();
  }

  if (tid == 0) {
    const float pos = sim[base + label];
    float mx = pos;
#pragma unroll
    for (int i = 0; i < NHARD; ++i) mx = fmaxf(mx, cand[i]);
    float sum = __expf(pos - mx);
#pragma unroll
    for (int i = 0; i < NHARD; ++i) sum += __expf(cand[i] - mx);
    nce_r[b] = mx + __logf(sum) - pos;  // logsumexp(logits) - pos
  }
}

// ---------------- cross-view alignment per row -----------------------------
__global__ void align_kernel(const float* __restrict__ eb,
                             const float* __restrict__ qn,   // normalized embeddings
                             const float* __restrict__ sat,
                             const int* __restrict__ labels,
                             float* __restrict__ align_r) {
  __shared__ float4 s[128];
  const int b = blockIdx.x;
  const int tid = threadIdx.x;
  const long long pbase = (long long)labels[b] * DE;
  const long long qbase = (long long)b * DE;

  float pp = 0.0f, dp = 0.0f, sp = 0.0f, ss = 0.0f;
  for (int i = tid; i < DE; i += 128) {
    const float p = eb[pbase + i];
    const float d = qn[qbase + i];
    const float sv = sat[qbase + i];
    pp += p * p; dp += d * p; sp += sv * p; ss += sv * sv;
  }
  s[tid] = make_float4(pp, dp, sp, ss);
  __syncthreads();
  for (int st = 64; st >= 1; st >>= 1) {
    if (tid < st) {
      float4 o = s[tid + st];
      s[tid].x += o.x; s[tid].y += o.y; s[tid].z += o.z; s[tid].w += o.w;
    }
    __syncthreads();
  }
  if (tid == 0) {
    const float invP = 1.0f / fmaxf(sqrtf(s[0].x), 1e-12f);
    const float invS = 1.0f / fmaxf(sqrtf(s[0].w), 1e-12f);
    align_r[b] = 0.5f * ((1.0f - s[0].y * invP) + (1.0f - s[0].z * invP * invS));
  }
}

// ---------------- final deterministic scalar reduction ---------------------
__global__ void final_reduce_kernel(const float* __restrict__ nce_r,
                                    const float* __restrict__ align_r,
                                    float* __restrict__ out2) {
  __shared__ float s[512];
  const int t = threadIdx.x;
  s[t] = nce_r[t];
  __syncthreads();
  for (int st = 256; st >= 1; st >>= 1) {
    if (t < st) s[t] += s[t + st];
    __syncthreads();
  }
  if (t == 0) out2[0] = s[0] / (float)BATCH;
  __syncthreads();
  s[t] = align_r[t];
  __syncthreads();
  for (int st = 256; st >= 1; st >>= 1) {
    if (t < st) s[t] += s[t + st];
    __syncthreads();
  }
  if (t == 0) out2[1] = s[0] / (float)BATCH;
}

// ---------------------------------------------------------------------------
extern "C" void kernel_launch(void* const* d_in, const int* in_sizes, int n_in,
                              void* d_out, int out_size, void* d_ws, size_t ws_size,
                              hipStream_t stream) {
  const float* part_features = (const float*)d_in[0];  // [512, 8, 256]
  const float* embeddings    = (const float*)d_in[1];  // [512, 512]
  const float* sat_emb       = (const float*)d_in[2];  // [512, 512]
  const float* part_bank     = (const float*)d_in[3];  // [20000, 8, 256]
  const float* embed_bank    = (const float*)d_in[4];  // [20000, 512]
  const int*   labels        = (const int*)d_in[5];    // [512]

  float* out       = (float*)d_out;
  float* out_part  = out;                       // P_ELEMS
  float* out_embed = out + P_ELEMS;             // E_ELEMS
  float* out_loss  = out + P_ELEMS + E_ELEMS;   // [nce, align]

  // workspace layout (floats)
  float* ws      = (float*)d_ws;
  float* qn      = ws;                                   // 512*512
  float* invn    = qn + (long long)BATCH * DE;           // 20000
  float* sim     = invn + CLS;                           // 512*20000
  float* nce_r   = sim + (long long)BATCH * CLS;         // 512
  float* align_r = nce_r + BATCH;                        // 512

  // 1) stream banks -> output (dominant HBM traffic, ~400 MB)
  copy_f4_kernel<<<4096, 256, 0, stream>>>((const float4*)part_bank,
                                           (float4*)out_part, P_ELEMS / 4);
  copy_f4_kernel<<<4096, 256, 0, stream>>>((const float4*)embed_bank,
                                           (float4*)out_embed, E_ELEMS / 4);
  // 2) EMA scatter of the 512 labeled rows (overwrites copied rows)
  ema_update_kernel<<<BATCH, 256, 0, stream>>>(part_features, embeddings,
                                               part_bank, embed_bank, labels,
                                               out_part, out_embed);
  // 3) normalize query embeddings
  row_normalize_kernel<<<BATCH, 256, 0, stream>>>(embeddings, qn, DE);
  // 4) per-class inverse norms of the updated embed bank
  proto_invnorm_kernel<<<CLS, 256, 0, stream>>>(out_embed, invn);
  // 5) similarity GEMM via f32 WMMA + async global->LDS staging
  dim3 ggrid((CLS / 16 + 3) / 4, BATCH / 16);  // (313, 32)
  sim_gemm_wmma_kernel<<<ggrid, 128, 0, stream>>>(qn, out_embed, invn, sim);
  // 6) hard-negative top-16 + logsumexp per row
  topk_lse_kernel<<<BATCH, 256, 0, stream>>>(sim, labels, nce_r);
  // 7) alignment loss per row
  align_kernel<<<BATCH, 128, 0, stream>>>(out_embed, qn, sat_emb, labels, align_r);
  // 8) deterministic scalar means
  final_reduce_kernel<<<1, 512, 0, stream>>>(nce_r, align_r, out_loss);
}